// NMS_78907139162555
// MI455X (gfx1250) — compile-verified
//
#include <hip/hip_runtime.h>

#define NUM_CLASSES 80
#define IOU_THR 0.5f
#define SCORE_THR 0.25f
#define MAX_BB 100
#define NEGV -1000000000.0f
#define NANCH 25200
#define NPAD 25600
#define TPB 1024
#define PERT 25   /* NPAD / TPB */
#define TDM_FLOATS 12800          /* first half of scores via TDM */

typedef __attribute__((ext_vector_type(2))) float v2f;
typedef __attribute__((ext_vector_type(8))) float v8f;
typedef __attribute__((ext_vector_type(4))) unsigned int v4u;
typedef __attribute__((ext_vector_type(4))) int v4i;
typedef __attribute__((ext_vector_type(8))) int v8i;

// ---------------------------------------------------------------------------
// Kernel 1: per-anchor class argmax, score = p * argmax_idx, threshold, and
// compaction of the 3 heads into contiguous padded (boxes, scores) workspace.
// Pure streaming: ~68 MB read @ 23.3 TB/s ≈ 3 us.
// ---------------------------------------------------------------------------
__global__ void score_kernel(
    const float* __restrict__ b20, const float* __restrict__ p20, const float* __restrict__ c20,
    const float* __restrict__ b40, const float* __restrict__ p40, const float* __restrict__ c40,
    const float* __restrict__ b80, const float* __restrict__ p80, const float* __restrict__ c80,
    float* __restrict__ boxes, float* __restrict__ scores, int B)
{
  int gid = blockIdx.x * blockDim.x + threadIdx.x;
  if (gid >= B * NPAD) return;
  int b = gid / NPAD;
  int slot = gid - b * NPAD;

  float4 box = make_float4(0.f, 0.f, 0.f, 0.f);
  float sc = NEGV;

  if (slot < NANCH) {
    const float *bp, *pp, *cp;
    int li, hn;
    if (slot < 1200)      { li = slot;        hn = 1200;  bp = b20; pp = p20; cp = c20; }
    else if (slot < 6000) { li = slot - 1200; hn = 4800;  bp = b40; pp = p40; cp = c40; }
    else                  { li = slot - 6000; hn = 19200; bp = b80; pp = p80; cp = c80; }
    long base = (long)b * hn + li;
    box = ((const float4*)bp)[base];
    float pobj = pp[base];
    const float* crow = cp + base * NUM_CLASSES;
    float mx = crow[0];
    int   mi = 0;
    #pragma unroll 4
    for (int j = 1; j < NUM_CLASSES; ++j) {
      float v = crow[j];
      if (v > mx) { mx = v; mi = j; }   // strict > => first-occurrence like jnp.argmax
    }
    float s = pobj * (float)mi;
    sc = (s > SCORE_THR) ? s : NEGV;
  }
  ((float4*)boxes)[gid] = box;
  scores[gid] = sc;
}

// Issue one TDM descriptor: copy `nfloats` f32 from gaddr -> LDS byte addr.
// 2D D#: data_size=4B, tensor_dim0 = tile_dim0 = nfloats, 1 row.
__device__ __forceinline__ void tdm_load_scores(unsigned lds_addr,
                                                unsigned long long gaddr,
                                                unsigned nfloats) {
  v4u g0;
  g0.x = 1u;                                        // count=1, is_restore=0, gather=0
  g0.y = lds_addr;                                  // lds_addr (bytes)
  g0.z = (unsigned)(gaddr & 0xffffffffull);         // global_addr[31:0]
  g0.w = (unsigned)((gaddr >> 32) & 0x01ffffffu)    // global_addr[56:32]
         | 0x80000000u;                             // type = 2 ("image")
  v8i g1;
  g1[0] = (int)(2u << 16);                          // wg_mask=0, data_size=2 (4B)
  g1[1] = (int)((nfloats & 0xffffu) << 16);         // tensor_dim0[15:0] @ bits 63:48
  g1[2] = (int)((nfloats >> 16) | (1u << 16));      // tensor_dim0[31:16] | tensor_dim1=1
  g1[3] = (int)((nfloats & 0xffffu) << 16);         // tile_dim0 @ bits 127:112
  g1[4] = 1;                                        // tile_dim1=1, tile_dim2=0
  g1[5] = (int)nfloats;                             // tensor_dim0_stride[31:0]
  g1[6] = 0;                                        // stride hi / dim1_stride lo
  g1[7] = 0;
  v4i z4 = {0, 0, 0, 0};
#if __has_include(<hip/amd_detail/amd_gfx1250_TDM.h>)
  v8i z8 = {0, 0, 0, 0, 0, 0, 0, 0};
  __builtin_amdgcn_tensor_load_to_lds(g0, g1, z4, z4, z8, 0);   // clang-23 6-arg form
#else
  __builtin_amdgcn_tensor_load_to_lds(g0, g1, z4, z4, 0);       // ROCm 7.2 5-arg form
#endif
}

// ---------------------------------------------------------------------------
// Kernel 2: one workgroup (1024 threads = 32 waves) per batch image.
// Scores live in LDS (100 KB of the 320 KB WGP LDS): lower half filled by the
// Tensor Data Mover (TENSORcnt), upper half by async global->LDS (ASYNCcnt).
// 100 serial rounds of wave32-shfl argmax -> broadcast -> IoU suppression.
// Valid-count reduced on the matrix pipe (v_wmma_f32_16x16x4_f32 x2).
// ---------------------------------------------------------------------------
__global__ __launch_bounds__(TPB, 1) void nms_kernel(
    const float* __restrict__ boxes, const float* __restrict__ scores,
    float* __restrict__ pred, int* __restrict__ validc)
{
  __shared__ float s_score[NPAD];
  __shared__ float s_wval[32];
  __shared__ int   s_widx[32];
  __shared__ float s_box[4];
  __shared__ float s_flags[128];

  const int t = threadIdx.x;
  const int b = blockIdx.x;
  const float*  sc_g = scores + (size_t)b * NPAD;
  const float4* bx_g = (const float4*)boxes + (size_t)b * NPAD;

  if (t < 128) s_flags[t] = 0.0f;

  unsigned lds0 = (unsigned)(size_t)(&s_score[0]);   // generic ptr low 32b == LDS offset

  // ---- TDM: DMA scores[0 .. 12800) into LDS (one descriptor, wave 0) ----
  if (t < 32) {   // wave-uniform: threads 0-31 are exactly wave 0 (wave32)
    tdm_load_scores(lds0, (unsigned long long)(size_t)sc_g, TDM_FLOATS);
    __builtin_amdgcn_s_wait_tensorcnt(0);
  }

  // ---- async global->LDS: scores[12800 .. 25600) (b128, ASYNCcnt) ----
  for (int i = (TDM_FLOATS / 4) + t; i < NPAD / 4; i += TPB) {
    unsigned           l = lds0 + (unsigned)(i * 16);
    unsigned long long g = (unsigned long long)(size_t)sc_g + (unsigned long long)(i * 16);
    asm volatile("global_load_async_to_lds_b128 %0, %1, off" :: "v"(l), "v"(g) : "memory");
  }

  // ---- warm L2 with this image's boxes while the fills are in flight ----
  #pragma unroll
  for (int k = 0; k < PERT; ++k)
    __builtin_prefetch((const void*)(bx_g + k * TPB + t), 0, 1);

  asm volatile("s_wait_asynccnt 0" ::: "memory");
  __syncthreads();

  for (int it = 0; it < MAX_BB; ++it) {
    // ---- phase A: local argmax over my 25 strided slots (bank-conflict free)
    float bv = -3.0e38f;
    int   bi = 0x7fffffff;
    #pragma unroll
    for (int k = 0; k < PERT; ++k) {
      int slot = k * TPB + t;
      float v = s_score[slot];
      if (v > bv || (v == bv && slot < bi)) { bv = v; bi = slot; }
    }
    // wave32 shuffle reduction (lower index wins ties, matching jnp.argmax)
    #pragma unroll
    for (int m = 16; m >= 1; m >>= 1) {
      float ov = __shfl_xor(bv, m, 32);
      int   oi = __shfl_xor(bi, m, 32);
      if (ov > bv || (ov == bv && oi < bi)) { bv = ov; bi = oi; }
    }
    if ((t & 31) == 0) { s_wval[t >> 5] = bv; s_widx[t >> 5] = bi; }
    __syncthreads();

    // ---- phase B/C: cross-wave reduce on wave 0, broadcast winner
    if (t < 32) {
      float v  = s_wval[t];
      int   i2 = s_widx[t];
      #pragma unroll
      for (int m = 16; m >= 1; m >>= 1) {
        float ov = __shfl_xor(v, m, 32);
        int   oi = __shfl_xor(i2, m, 32);
        if (ov > v || (ov == v && oi < i2)) { v = ov; i2 = oi; }
      }
      if (t == 0) {
        float4 bb = bx_g[i2];
        s_box[0] = bb.x; s_box[1] = bb.y; s_box[2] = bb.z; s_box[3] = bb.w;
        s_score[i2] = NEGV;                       // sc.at[idx].set(NEG)
        bool valid = v > (NEGV * 0.5f);
        float o0 = valid ? fminf(fmaxf(bb.x, 0.f), 1.f) : 0.f;
        float o1 = valid ? fminf(fmaxf(bb.y, 0.f), 1.f) : 0.f;
        float o2 = valid ? fminf(fmaxf(bb.z, 0.f), 1.f) : 0.f;
        float o3 = valid ? fminf(fmaxf(bb.w, 0.f), 1.f) : 0.f;
        float o4 = valid ? v : 0.f;
        size_t ob = ((size_t)b * MAX_BB + (size_t)it) * 6;
        pred[ob + 0] = o0; pred[ob + 1] = o1; pred[ob + 2] = o2;
        pred[ob + 3] = o3; pred[ob + 4] = o4; pred[ob + 5] = 0.f;
        s_flags[it] = valid ? 1.0f : 0.0f;
      }
    }
    __syncthreads();

    // ---- phase D: IoU suppression against my 25 slots (boxes from L2)
    // unroll-5 caps in-flight b128 loads at 5 (20 VGPRs) to avoid spills
    // under the 128-VGPR/32-wave budget, while keeping enough MLP for L2.
    float y1 = s_box[0], x1 = s_box[1], y2 = s_box[2], x2 = s_box[3];
    float area = (y2 - y1) * (x2 - x1);
    #pragma unroll 5
    for (int k = 0; k < PERT; ++k) {
      int slot = k * TPB + t;
      float4 q = bx_g[slot];                     // q = (Y1, X1, Y2, X2)
      float ih = fmaxf(fminf(y2, q.z) - fmaxf(y1, q.x), 0.f);
      float iw = fmaxf(fminf(x2, q.w) - fmaxf(x1, q.y), 0.f);
      float inter = ih * iw;
      float uni = area + (q.z - q.x) * (q.w - q.y) - inter;
      float iou = (uni > 0.f) ? (inter / uni) : 0.f;
      if (iou > IOU_THR) s_score[slot] = NEGV;
    }
    __syncthreads();
  }

  // ---- valid count via matrix pipe: rowsum(flags 16x4) * ones, twice ----
  if (t < 32) {
    int m   = t & 15;
    int hik = (t >> 4) & 1;      // A layout: lanes 0-15 hold K=0,1 ; lanes 16-31 hold K=2,3
    v2f a1, a2, bones;
    a1.x = s_flags[m * 4 + (hik ? 2 : 0)];
    a1.y = s_flags[m * 4 + (hik ? 3 : 1)];
    a2.x = s_flags[64 + m * 4 + (hik ? 2 : 0)];
    a2.y = s_flags[64 + m * 4 + (hik ? 3 : 1)];
    bones.x = 1.0f; bones.y = 1.0f;
    v8f c = {0.f, 0.f, 0.f, 0.f, 0.f, 0.f, 0.f, 0.f};
    v8f d = __builtin_amdgcn_wmma_f32_16x16x4_f32(false, a1, false, bones, (short)0, c, false, false);
    d     = __builtin_amdgcn_wmma_f32_16x16x4_f32(false, a2, false, bones, (short)0, d, false, false);
    // D[m][n]: lane n holds column n; VGPR r = row r (lanes 0-15) / row 8+r (lanes 16-31).
    float partial = d[0] + d[1] + d[2] + d[3] + d[4] + d[5] + d[6] + d[7]; // column-n partial
    float other   = __shfl_xor(partial, 16, 32);   // lane0 gets rows 8..15 of column 0
    if (t == 0) validc[b] = (int)(partial + other + 0.5f);
  }
}

extern "C" void kernel_launch(void* const* d_in, const int* in_sizes, int n_in,
                              void* d_out, int out_size, void* d_ws, size_t ws_size,
                              hipStream_t stream) {
  (void)n_in; (void)out_size; (void)ws_size;
  const float* b20 = (const float*)d_in[0];
  const float* p20 = (const float*)d_in[1];
  const float* c20 = (const float*)d_in[2];
  const float* b40 = (const float*)d_in[3];
  const float* p40 = (const float*)d_in[4];
  const float* c40 = (const float*)d_in[5];
  const float* b80 = (const float*)d_in[6];
  const float* p80 = (const float*)d_in[7];
  const float* c80 = (const float*)d_in[8];

  int B = in_sizes[0] / (20 * 20 * 3 * 4);   // = 8 for the reference

  float* boxes  = (float*)d_ws;                         // B*NPAD*4 floats
  float* scores = boxes + (size_t)B * NPAD * 4;         // B*NPAD floats
  float* pred   = (float*)d_out;                        // B*100*6 floats
  int*   validc = (int*)((float*)d_out + (size_t)B * MAX_BB * 6);

  int total = B * NPAD;
  score_kernel<<<(total + 255) / 256, 256, 0, stream>>>(
      b20, p20, c20, b40, p40, c40, b80, p80, c80, boxes, scores, B);
  nms_kernel<<<B, TPB, 0, stream>>>(boxes, scores, pred, validc);
}